// diff_feat_encoder_44822278701373
// MI455X (gfx1250) — compile-verified
//
#include <hip/hip_runtime.h>
#include <stdint.h>

// ---------------------------------------------------------------------------
// VN graph encoder, fp32 end-to-end using CDNA5 V_WMMA_F32_16X16X4_F32.
// B=16, N=2048, K=32, H=32, OUT=64.
// ---------------------------------------------------------------------------

#define BATCH 16
#define NPTS  2048
#define KNEI  32
#define HCH   32
#define EPSF  1e-6f

typedef __attribute__((ext_vector_type(2))) float v2f;
typedef __attribute__((ext_vector_type(8))) float v8f;

// D = A(16x4) * B(4x16) + C(16x16), all fp32, wave32.
// 8-arg pattern per probe: (neg_a, A, neg_b, B, c_mod, C, reuse_a, reuse_b)
__device__ __forceinline__ v8f wmma4f32(v2f a, v2f b, v8f c) {
    return __builtin_amdgcn_wmma_f32_16x16x4_f32(false, a, false, b,
                                                 (short)0, c, false, false);
}

// ===========================================================================
// Kernel A: stage0 edge features + WMMA pool + argmax-over-K gather.
// Block = 256 threads (8 waves) = 4 points. feat0 LDS: [c=32][pt*64 + 2k+dd]
// ===========================================================================
#define FS 264   // 4*64 cols + 8 pad (floats)

__global__ __launch_bounds__(256)
void vn_stage0_pool_kernel(const float* __restrict__ x,
                           const unsigned char* __restrict__ mask,  // jnp bool (1B)
                           const int* __restrict__ nidx,
                           const float* __restrict__ Wf0,
                           const float* __restrict__ Wd0,
                           const float* __restrict__ Wpool,
                           float* __restrict__ pooled /* [B*N][32][2] */) {
    __shared__ float feat[HCH * FS];
    __shared__ float wf0[64];
    __shared__ float wd0[64];

    const int tid = threadIdx.x;
    if (tid < 64)       wf0[tid]      = Wf0[tid];
    else if (tid < 128) wd0[tid - 64] = Wd0[tid - 64];

    const int p0 = blockIdx.x * 4;            // first global point id

    // ---- phase 1: stage0 (2 -> 32 channels, D=2) into LDS ----
    {
        const int pt  = tid >> 6;             // 0..3
        const int k   = (tid >> 1) & 31;      // 0..31
        const int ch0 = (tid & 1) * 16;       // channel half per thread
        const int pid = p0 + pt;
        const int b   = pid >> 11;
        const int n   = pid & 2047;
        const int e   = (b * NPTS + n) * KNEI + k;
        const int j   = nidx[e];
        const float m = mask[e] ? 1.0f : 0.0f;
        const float xn0 = m * x[(b * NPTS + j) * 2 + 0];
        const float xn1 = m * x[(b * NPTS + j) * 2 + 1];
        const float xs0 = x[(b * NPTS + n) * 2 + 0];
        const float xs1 = x[(b * NPTS + n) * 2 + 1];
        __syncthreads();                      // wf0/wd0 ready
        const int colbase = pt * 64 + 2 * k;
        #pragma unroll 4
        for (int c = ch0; c < ch0 + 16; ++c) {
            const float fa = wf0[2 * c], fb = wf0[2 * c + 1];
            const float da = wd0[2 * c], db = wd0[2 * c + 1];
            const float pA = fa * xn0 + fb * xs0;   // dd = 0
            const float pB = fa * xn1 + fb * xs1;   // dd = 1
            const float dA = da * xn0 + db * xs0;
            const float dB = da * xn1 + db * xs1;
            const float dot = pA * dA + pB * dB;
            const float dsq = dA * dA + dB * dB;
            const float s   = dot / (dsq + EPSF);
            feat[c * FS + colbase + 0] = (dot >= 0.f) ? pA : pA - s * dA;
            feat[c * FS + colbase + 1] = (dot >= 0.f) ? pB : pB - s * dB;
        }
    }
    __syncthreads();

    // ---- phase 2: pool GEMM (Wpool[32x32] x feat0[32x64/pt]) + argmax ----
    const int wv   = tid >> 5;      // wave id: pt = wv>>1, row-tile = wv&1
    const int lane = tid & 31;
    const int li   = lane & 15;
    const int hi   = lane >> 4;
    const int pt   = wv >> 1;
    const int rt   = wv & 1;

    const int pid = p0 + pt;
    const int b   = pid >> 11;
    const int n   = pid & 2047;

    float best[8];
    int   bk[8];
    #pragma unroll
    for (int i = 0; i < 8; ++i) { best[i] = -3.4e38f; bk[i] = 0; }

    const int arow = rt * 16 + li;
    for (int ct = 0; ct < 4; ++ct) {
        v8f acc = {};
        const int col = pt * 64 + ct * 16 + li;
        #pragma unroll
        for (int c = 0; c < HCH; c += 4) {
            const int ka = c + 2 * hi;
            v2f a, bb;
            a.x  = Wpool[arow * HCH + ka];
            a.y  = Wpool[arow * HCH + ka + 1];
            bb.x = feat[ka * FS + col];
            bb.y = feat[(ka + 1) * FS + col];
            acc = wmma4f32(a, bb, acc);
        }
        const int kk = ct * 8 + (li >> 1);
        #pragma unroll
        for (int i = 0; i < 8; ++i) {
            const int row = rt * 16 + i + 8 * hi;       // C-layout M
            const float f  = feat[row * FS + col];
            const float pd = f * acc[i];
            const float dot = pd + __shfl_xor(pd, 1, 32);  // sum over dd pair
            if (dot > best[i]) { best[i] = dot; bk[i] = kk; }
        }
    }
    // argmax reduce over the 8 k-pairs within each 16-lane half
    #pragma unroll
    for (int i = 0; i < 8; ++i) {
        #pragma unroll
        for (int mlev = 2; mlev <= 8; mlev <<= 1) {
            const float ob = __shfl_xor(best[i], mlev, 32);
            const int   ok = __shfl_xor(bk[i],   mlev, 32);
            if (ob > best[i] || (ob == best[i] && ok < bk[i])) {
                best[i] = ob; bk[i] = ok;
            }
        }
        if (li < 2) {   // dd = li; gather winning neighbor's feature
            const int c = rt * 16 + i + 8 * hi;
            const float v = feat[c * FS + pt * 64 + 2 * bk[i] + li];
            pooled[((b * NPTS + n) * HCH + c) * 2 + li] = v;
        }
    }
}

// ===========================================================================
// Kernel B: 4 VN linear+LReLU stages via WMMA. Block = 256 threads = 32 pts.
// LDS X layout: [channel][col], col = 2*pt_local + dd.
// ===========================================================================
#define XS 66    // 64 cols + 2 pad

template <int CIN, int COUT>
__device__ __forceinline__ void vn_stage(const float* __restrict__ Wf,
                                         const float* __restrict__ Wd,
                                         const float* __restrict__ Xin,
                                         float* __restrict__ Xout, int tid) {
    const int wv = tid >> 5, lane = tid & 31, li = lane & 15, hi = lane >> 4;
    const int ntile = (COUT >> 4) * 4;     // row tiles * col tiles
    for (int task = wv; task < ntile; task += 8) {
        const int rt = task >> 2, ct = task & 3;
        const int col  = ct * 16 + li;
        const int arow = rt * 16 + li;
        v8f accP = {}, accD = {};
        #pragma unroll
        for (int c = 0; c < CIN; c += 4) {
            const int ka = c + 2 * hi;
            v2f af, ad, bx;
            af.x = Wf[arow * CIN + ka];  af.y = Wf[arow * CIN + ka + 1];
            ad.x = Wd[arow * CIN + ka];  ad.y = Wd[arow * CIN + ka + 1];
            bx.x = Xin[ka * XS + col];   bx.y = Xin[(ka + 1) * XS + col];
            accP = wmma4f32(af, bx, accP);
            accD = wmma4f32(ad, bx, accD);
        }
        #pragma unroll
        for (int i = 0; i < 8; ++i) {
            const float p  = accP[i], d = accD[i];
            const float pd = p * d;
            const float dot = pd + __shfl_xor(pd, 1, 32);   // sum over dd
            const float d2  = d * d;
            const float dsq = d2 + __shfl_xor(d2, 1, 32);
            const float s   = dot / (dsq + EPSF);
            const float o   = (dot >= 0.f) ? p : p - s * d;
            Xout[(rt * 16 + i + 8 * hi) * XS + col] = o;
        }
    }
}

__global__ __launch_bounds__(256)
void vn_mlp_kernel(const float* __restrict__ pooled,
                   const float* __restrict__ Wf1, const float* __restrict__ Wd1,
                   const float* __restrict__ Wf2, const float* __restrict__ Wd2,
                   const float* __restrict__ Wf3, const float* __restrict__ Wd3,
                   const float* __restrict__ Wfo, const float* __restrict__ Wdo,
                   float* __restrict__ out /* [B][64][2][N] */) {
    __shared__ float Xa[64 * XS];
    __shared__ float Xb[64 * XS];

    const int tid = threadIdx.x;
    const int p0  = blockIdx.x * 32;
    const int b   = p0 >> 11;
    const int n0  = p0 & 2047;

    // load pooled [pt][c][dd] -> Xa[c][2*pt+dd]  (coalesced global reads)
    const float* pbase = pooled + (size_t)(b * NPTS + n0) * 64;
    for (int e = tid; e < 32 * 64; e += 256) {
        const int pt = e >> 6, cd = e & 63;
        Xa[(cd >> 1) * XS + 2 * pt + (cd & 1)] = pbase[e];
    }
    __syncthreads();

    vn_stage<32, 64>(Wf1, Wd1, Xa, Xb, tid);  __syncthreads();
    vn_stage<64, 64>(Wf2, Wd2, Xb, Xa, tid);  __syncthreads();
    vn_stage<64, 32>(Wf3, Wd3, Xa, Xb, tid);  __syncthreads();
    vn_stage<32, 64>(Wfo, Wdo, Xb, Xa, tid);  __syncthreads();

    // store Xa[o][2*nl+dd] -> out[((b*64+o)*2+dd)*N + n0+nl]  (32-wide coalesced)
    for (int e = tid; e < 64 * 64; e += 256) {
        const int o = e >> 6, dd = (e >> 5) & 1, nl = e & 31;
        out[((b * 64 + o) * 2 + dd) * NPTS + n0 + nl] = Xa[o * XS + 2 * nl + dd];
    }
}

// ===========================================================================
extern "C" void kernel_launch(void* const* d_in, const int* in_sizes, int n_in,
                              void* d_out, int out_size, void* d_ws, size_t ws_size,
                              hipStream_t stream) {
    (void)in_sizes; (void)n_in; (void)out_size; (void)ws_size;
    const float*         x     = (const float*)d_in[0];
    const unsigned char* mask  = (const unsigned char*)d_in[1];  // jnp bool = 1 byte
    const int*           nidx  = (const int*)d_in[2];
    const float* Wf0   = (const float*)d_in[3];
    const float* Wd0   = (const float*)d_in[4];
    const float* Wpool = (const float*)d_in[5];
    const float* Wf1   = (const float*)d_in[6];
    const float* Wd1   = (const float*)d_in[7];
    const float* Wf2   = (const float*)d_in[8];
    const float* Wd2   = (const float*)d_in[9];
    const float* Wf3   = (const float*)d_in[10];
    const float* Wd3   = (const float*)d_in[11];
    const float* Wfo   = (const float*)d_in[12];
    const float* Wdo   = (const float*)d_in[13];

    float* pooled = (float*)d_ws;            // [B*N][32][2] = 8 MB
    float* outp   = (float*)d_out;

    vn_stage0_pool_kernel<<<(BATCH * NPTS) / 4, 256, 0, stream>>>(
        x, mask, nidx, Wf0, Wd0, Wpool, pooled);

    vn_mlp_kernel<<<(BATCH * NPTS) / 32, 256, 0, stream>>>(
        pooled, Wf1, Wd1, Wf2, Wd2, Wf3, Wd3, Wfo, Wdo, outp);
}